// MSDeformAttn_412316860432
// MI455X (gfx1250) — compile-verified
//
#include <hip/hip_runtime.h>

// ---- Problem constants (fixed by the reference) ----
#define NB      2
#define LQ      5440
#define CH      256
#define MH      8
#define LV      4
#define PT      4
#define DH      32        // CH / MH
#define LEN_IN  5440      // 64*64 + 32*32 + 16*16 + 8*8
#define ROWS    (NB * LQ) // 10880 (== NB*LEN_IN too)

typedef __attribute__((ext_vector_type(2))) float v2f;
typedef __attribute__((ext_vector_type(8))) float v8f;

// ---------------------------------------------------------------------------
// FP32 WMMA GEMM: Out[Mrows x Ncols] = A[Mrows x K] * W[K x Ncols] + bias
// One wave32 per 16x16 output tile, K-loop in steps of 4 using
// V_WMMA_F32_16X16X4_F32. Mrows % 16 == 0, Ncols % 16 == 0, K % 4 == 0.
// ---------------------------------------------------------------------------
__global__ void gemm_f32_wmma_kernel(const float* __restrict__ A,
                                     const float* __restrict__ W,
                                     const float* __restrict__ bias,
                                     float* __restrict__ Out,
                                     int Mrows, int Ncols, int K)
{
    const int wave  = (int)((blockIdx.x * blockDim.x + threadIdx.x) >> 5);
    const int lane  = (int)(threadIdx.x & 31);
    const int tilesN = Ncols >> 4;
    const int tileM  = wave / tilesN;
    const int tileN  = wave - tileM * tilesN;
    if (tileM * 16 >= Mrows) return;          // wave-uniform guard (EXEC stays full)

    const int half = lane >> 4;               // 0: K={0,1}, 1: K={2,3}
    const int l16  = lane & 15;

    // A fragment source: row = tileM*16 + l16, contiguous pair at k0 + 2*half
    const float* __restrict__ arow = A + (size_t)(tileM * 16 + l16) * (size_t)K + (size_t)(half * 2);
    // B fragment source: col = tileN*16 + l16, rows k0 + 2*half + {0,1}
    const int bcol = tileN * 16 + l16;
    const float* __restrict__ bptr = W + (size_t)(half * 2) * (size_t)Ncols + (size_t)bcol;

    v8f acc = {};
    #pragma unroll 4
    for (int k0 = 0; k0 < K; k0 += 4) {
        __builtin_prefetch(arow + k0 + 32, 0, 1);   // global_prefetch_b8
        v2f a, b;
        a.x = arow[k0];
        a.y = arow[k0 + 1];
        b.x = bptr[(size_t)k0 * (size_t)Ncols];
        b.y = bptr[(size_t)(k0 + 1) * (size_t)Ncols];
        // 8 args: (neg_a, A, neg_b, B, c_mod, C, reuse_a, reuse_b)
        acc = __builtin_amdgcn_wmma_f32_16x16x4_f32(false, a, false, b,
                                                    (short)0, acc, false, false);
    }

    const float bv = bias[bcol];
    float* __restrict__ orow = Out + (size_t)(tileM * 16 + half * 8) * (size_t)Ncols + (size_t)bcol;
    #pragma unroll
    for (int r = 0; r < 8; ++r)
        orow[(size_t)r * (size_t)Ncols] = acc[r] + bv;
}

// ---------------------------------------------------------------------------
// Softmax over the last 16 (L*P) logits, in place. attn layout: (ROWS, M*16),
// thread t handles (row, m) with base offset t*16.
// ---------------------------------------------------------------------------
__global__ void softmax16_kernel(float* __restrict__ attn, int total)
{
    const int t = (int)(blockIdx.x * blockDim.x + threadIdx.x);
    if (t >= total) return;
    float* __restrict__ p = attn + (size_t)t * 16;
    float mx = p[0];
    #pragma unroll
    for (int i = 1; i < 16; ++i) mx = fmaxf(mx, p[i]);
    float e[16];
    float s = 0.f;
    #pragma unroll
    for (int i = 0; i < 16; ++i) { e[i] = __expf(p[i] - mx); s += e[i]; }
    const float inv = 1.f / s;
    #pragma unroll
    for (int i = 0; i < 16; ++i) p[i] = e[i] * inv;
}

// ---------------------------------------------------------------------------
// Deformable sampling. One wave32 per (n, q, m); lane = channel d (D == 32).
// value layout: (N, LEN_IN, M, D)  -> lane-contiguous 128B gathers.
// offs  layout: (ROWS, M, L, P, 2) -> base wid*32
// attn  layout: (ROWS, M, 16)      -> base wid*16
// out   layout: (ROWS, M, D)       == (N, LQ, C)
// ---------------------------------------------------------------------------
__device__ __forceinline__ float msda_fetch(const float* __restrict__ v, size_t lvlbase,
                                            int H, int W, int yi, int xi, int m, int d)
{
    const bool ok = (xi >= 0) & (xi < W) & (yi >= 0) & (yi < H);
    int yc = yi < 0 ? 0 : (yi > H - 1 ? H - 1 : yi);
    int xc = xi < 0 ? 0 : (xi > W - 1 ? W - 1 : xi);
    const size_t pos = lvlbase + (size_t)yc * (size_t)W + (size_t)xc;
    const float val = v[(pos * MH + (size_t)m) * DH + (size_t)d];
    return ok ? val : 0.f;
}

__global__ void msda_sample_kernel(const float* __restrict__ value,
                                   const float* __restrict__ offs,
                                   const float* __restrict__ attn,
                                   const float* __restrict__ refp,
                                   float* __restrict__ out)
{
    const int wid = (int)((blockIdx.x * blockDim.x + threadIdx.x) >> 5);
    const int d   = (int)(threadIdx.x & 31);
    if (wid >= ROWS * MH) return;            // wave-uniform

    const int m  = wid & (MH - 1);
    const int nq = wid >> 3;                 // n*LQ + q
    const int n  = nq / LQ;

    const int   Hs[LV]     = {64, 32, 16, 8};
    const int   Wss[LV]    = {64, 32, 16, 8};
    const int   starts[LV] = {0, 4096, 5120, 5376};

    const float* __restrict__ aP = attn + (size_t)wid * 16;
    const float* __restrict__ oP = offs + (size_t)wid * 32;
    const float* __restrict__ rP = refp + (size_t)nq * (LV * 2);
    const size_t vbase_n = (size_t)n * LEN_IN;

    float acc = 0.f;
    #pragma unroll
    for (int l = 0; l < LV; ++l) {
        const int   H  = Hs[l], W = Wss[l];
        const float fw = (float)W, fh = (float)H;
        const float rx = rP[l * 2 + 0];
        const float ry = rP[l * 2 + 1];
        const size_t lvlbase = vbase_n + (size_t)starts[l];
        #pragma unroll
        for (int p = 0; p < PT; ++p) {
            const float aw = aP[l * PT + p];
            const float ox = oP[(l * PT + p) * 2 + 0];
            const float oy = oP[(l * PT + p) * 2 + 1];
            // loc = ref + off/normalizer ; grid = 2*loc - 1 ; clip to [-2,2]
            float gx = 2.f * (rx + ox / fw) - 1.f;
            float gy = 2.f * (ry + oy / fh) - 1.f;
            gx = fminf(fmaxf(gx, -2.f), 2.f);
            gy = fminf(fmaxf(gy, -2.f), 2.f);
            // align_corners = False
            const float x = (gx + 1.f) * 0.5f * fw - 0.5f;
            const float y = (gy + 1.f) * 0.5f * fh - 0.5f;
            const float x0f = floorf(x), y0f = floorf(y);
            const float wx = x - x0f, wy = y - y0f;
            const int x0 = (int)x0f, y0 = (int)y0f;

            const float v00 = msda_fetch(value, lvlbase, H, W, y0,     x0,     m, d);
            const float v01 = msda_fetch(value, lvlbase, H, W, y0,     x0 + 1, m, d);
            const float v10 = msda_fetch(value, lvlbase, H, W, y0 + 1, x0,     m, d);
            const float v11 = msda_fetch(value, lvlbase, H, W, y0 + 1, x0 + 1, m, d);

            const float s = v00 * (1.f - wx) * (1.f - wy)
                          + v01 * wx         * (1.f - wy)
                          + v10 * (1.f - wx) * wy
                          + v11 * wx         * wy;
            acc += aw * s;
        }
    }
    out[(size_t)wid * DH + (size_t)d] = acc;
}

// ---------------------------------------------------------------------------
// Host launcher
// ---------------------------------------------------------------------------
extern "C" void kernel_launch(void* const* d_in, const int* in_sizes, int n_in,
                              void* d_out, int out_size, void* d_ws, size_t ws_size,
                              hipStream_t stream)
{
    (void)in_sizes; (void)n_in; (void)out_size; (void)ws_size;

    const float* query  = (const float*)d_in[0];   // (N, LQ, C)
    const float* refp   = (const float*)d_in[1];   // (N, LQ, L, 2)
    const float* inflat = (const float*)d_in[2];   // (N, LEN_IN, C)
    // d_in[3], d_in[4]: spatial shapes / level starts — compile-time constants here
    const float* W_off  = (const float*)d_in[5];   // (C, 256)
    const float* b_off  = (const float*)d_in[6];   // (256,)
    const float* W_attn = (const float*)d_in[7];   // (C, 128)
    const float* b_attn = (const float*)d_in[8];   // (128,)
    const float* W_val  = (const float*)d_in[9];   // (C, C)
    const float* b_val  = (const float*)d_in[10];  // (C,)
    const float* W_out  = (const float*)d_in[11];  // (C, C)
    const float* b_out  = (const float*)d_in[12];  // (C,)

    float* out = (float*)d_out;                    // (N, LQ, C)
    float* ws  = (float*)d_ws;

    // Workspace layout (floats)
    float* val_ws  = ws;                                       // ROWS * 256
    float* off_ws  = val_ws  + (size_t)ROWS * CH;              // ROWS * 256
    float* attn_ws = off_ws  + (size_t)ROWS * CH;              // ROWS * 128
    float* samp_ws = attn_ws + (size_t)ROWS * (MH * LV * PT);  // ROWS * 256

    auto launch_gemm = [&](const float* A, const float* W, const float* b,
                           float* O, int Mr, int Nc) {
        const int tiles   = (Mr / 16) * (Nc / 16);
        const int threads = tiles * 32;
        const int blocks  = (threads + 255) / 256;
        gemm_f32_wmma_kernel<<<blocks, 256, 0, stream>>>(A, W, b, O, Mr, Nc, CH);
    };

    // 1) value = input_flatten @ W_val + b_val          (10880 x 256)
    launch_gemm(inflat, W_val, b_val, val_ws, ROWS, CH);
    // 2) offsets = query @ W_off + b_off                (10880 x 256)
    launch_gemm(query, W_off, b_off, off_ws, ROWS, CH);
    // 3) attn logits = query @ W_attn + b_attn          (10880 x 128)
    launch_gemm(query, W_attn, b_attn, attn_ws, ROWS, MH * LV * PT);
    // 4) softmax over L*P per (n,q,m)
    softmax16_kernel<<<(ROWS * MH + 255) / 256, 256, 0, stream>>>(attn_ws, ROWS * MH);
    // 5) deformable bilinear sampling + head aggregation (wave per (n,q,m))
    msda_sample_kernel<<<(ROWS * MH * 32 + 255) / 256, 256, 0, stream>>>(
        val_ws, off_ws, attn_ws, refp, samp_ws);
    // 6) out = sampled @ W_out + b_out                  (10880 x 256)
    launch_gemm(samp_ws, W_out, b_out, out, ROWS, CH);
}